// ScaledDotProductAttention_15616501088388
// MI455X (gfx1250) — compile-verified
//
#include <hip/hip_runtime.h>
#include <hip/hip_bf16.h>

// Scaled dot-product attention forward, f32, with bias, returning (out, attn).
// B=2, H=16, S=2048, D=64.  d_out = [out (B*H*S*D)] ++ [attn (B*H*S*S)].
//
// One workgroup (256 threads / 8 wave32) handles a 16-row Q tile of one (b,h).
// Matmuls use V_WMMA_F32_16X16X4_F32 (CDNA5 f32 matrix op), accumulator
// initialized from bias so scores = (Q/8)*K^T + bias comes out of the WMMA
// chain directly.

typedef __attribute__((ext_vector_type(2))) float v2f;
typedef __attribute__((ext_vector_type(8))) float v8f;

#define SEQ     2048
#define DHEAD   64
#define NTILES  (SEQ / 16)      // 128 key tiles per q-row-tile
#define WAVES   8
#define NTHR    256
#define PS      (SEQ + 4)       // padded LDS row stride (banks: 2052 % 64 == 4)

__launch_bounds__(NTHR)
__global__ void attn_fwd_kernel(const float* __restrict__ q,
                                const float* __restrict__ k,
                                const float* __restrict__ v,
                                const float* __restrict__ bias,
                                float* __restrict__ out,
                                float* __restrict__ attn)
{
    __shared__ float smem[16 * PS];   // 16 x 2048 scores/probs (padded rows)
    __shared__ float red[NTHR];       // softmax reduction scratch
    __shared__ float invs[16];        // per-row 1/rowsum

    const int bid = blockIdx.x;
    const int bh  = bid >> 7;         // 0..31 = flattened (b,h)
    const int q0  = (bid & 127) * 16; // first q row of this tile

    const float* Q  = q    + (size_t)bh * SEQ * DHEAD;
    const float* K  = k    + (size_t)bh * SEQ * DHEAD;
    const float* V  = v    + (size_t)bh * SEQ * DHEAD;
    const float* Bb = bias + (size_t)bh * SEQ * SEQ;
    float* Ob = out  + (size_t)bh * SEQ * DHEAD;
    float* Ab = attn + (size_t)bh * SEQ * SEQ;

    const int tid  = threadIdx.x;
    const int wave = tid >> 5;
    const int lane = tid & 31;
    const int l    = lane & 15;       // column / M-row selector
    const int hi   = lane >> 4;       // which K-pair of the x4 chunk

    // ---- Q fragment in 16x4 f32 A-layout, pre-scaled by 1/sqrt(D)=1/8 ----
    // lane(l,hi) holds Q[q0+l][4*kc + 2*hi + {0,1}]
    v2f aq[16];
    {
        const float* qrow = Q + (size_t)(q0 + l) * DHEAD + hi * 2;
#pragma unroll
        for (int kc = 0; kc < 16; ++kc) {
            v2f t = *reinterpret_cast<const v2f*>(qrow + kc * 4);
            aq[kc].x = t.x * 0.125f;
            aq[kc].y = t.y * 0.125f;
        }
    }

    // ---- scores tile: (Q/8) K^T + bias via WMMA f32 16x16x4 chain ----
    for (int nt = wave; nt < NTILES; nt += WAVES) {
        const int n0 = nt * 16;

        // C init = bias in C-layout: vgpr r -> (row = r + 8*hi, col = l)
        v8f acc;
        const float* bb = Bb + (size_t)(q0 + hi * 8) * SEQ + n0 + l;
#pragma unroll
        for (int r = 0; r < 8; ++r)
            acc[r] = bb[(size_t)r * SEQ];

        // B fragment: B[kk][n] = K[n0+n][kk]; lane(l,hi) loads
        // K[n0+l][4*kc + 2*hi + {0,1}]
        const float* krow = K + (size_t)(n0 + l) * DHEAD + hi * 2;
#pragma unroll
        for (int kc = 0; kc < 16; ++kc) {
            v2f bk = *reinterpret_cast<const v2f*>(krow + kc * 4);
            acc = __builtin_amdgcn_wmma_f32_16x16x4_f32(
                false, aq[kc], false, bk, (short)0, acc, false, false);
        }

        float* srow = &smem[(hi * 8) * PS + n0 + l];
#pragma unroll
        for (int r = 0; r < 8; ++r)
            srow[r * PS] = acc[r];
    }
    __syncthreads();

    // ---- row softmax over 2048 columns: 16 threads per row ----
    {
        const int r  = tid >> 4;      // row 0..15
        const int c0 = tid & 15;      // column phase

        float pmax = -1e30f;
#pragma unroll 8
        for (int c = c0; c < SEQ; c += 16)
            pmax = fmaxf(pmax, smem[r * PS + c]);
        red[tid] = pmax;
        __syncthreads();
        float rmax = -1e30f;
#pragma unroll
        for (int i = 0; i < 16; ++i)
            rmax = fmaxf(rmax, red[(r << 4) + i]);
        __syncthreads();

        float psum = 0.f;
#pragma unroll 8
        for (int c = c0; c < SEQ; c += 16) {
            float e = __expf(smem[r * PS + c] - rmax);
            smem[r * PS + c] = e;
            psum += e;
        }
        red[tid] = psum;
        __syncthreads();
        float rsum = 0.f;
#pragma unroll
        for (int i = 0; i < 16; ++i)
            rsum += red[(r << 4) + i];
        if (c0 == 0) invs[r] = 1.0f / rsum;
    }
    __syncthreads();

    // ---- normalize in LDS + coalesced attn store ----
    for (int idx = tid; idx < 16 * SEQ; idx += NTHR) {
        const int rr = idx >> 11;          // / 2048
        const int cc = idx & (SEQ - 1);
        float p = smem[rr * PS + cc] * invs[rr];
        smem[rr * PS + cc] = p;
        Ab[(size_t)(q0 + rr) * SEQ + cc] = p;
    }
    __syncthreads();

    // ---- out = P @ V, split-K WMMA: wave -> (dtile = w&3, khalf = w>>2) ----
    {
        const int d0    = (wave & 3) * 16;
        const int kbase = (wave >> 2) * (SEQ / 2);

        v8f acc = {0.f, 0.f, 0.f, 0.f, 0.f, 0.f, 0.f, 0.f};
        const float* prow = &smem[l * PS];      // A row M=l (padded stride)
#pragma unroll 4
        for (int kc = 0; kc < (SEQ / 2) / 4; ++kc) {
            const int kk = kbase + kc * 4 + hi * 2;
            v2f ap = *reinterpret_cast<const v2f*>(prow + kk);
            // B fragment: lane(l,hi) holds V[kk][d0+l], V[kk+1][d0+l]
            const float* vp = V + (size_t)kk * DHEAD + d0 + l;
            v2f bv;
            bv.x = vp[0];
            bv.y = vp[DHEAD];
            acc = __builtin_amdgcn_wmma_f32_16x16x4_f32(
                false, ap, false, bv, (short)0, acc, false, false);
        }
        __syncthreads();                        // all P reads complete
        // stash 16x16 partials in LDS (reuse scores buffer)
        float* pw = &smem[wave * 256];
#pragma unroll
        for (int r = 0; r < 8; ++r)
            pw[(r + hi * 8) * 16 + l] = acc[r];
    }
    __syncthreads();

    // ---- combine K-halves and store out tile (16 x 64) ----
    if (wave < 4) {
        const int d0 = wave * 16;
        const float* p0 = &smem[wave * 256];
        const float* p1 = &smem[(wave + 4) * 256];
        float* orow = Ob + (size_t)(q0 + hi * 8) * DHEAD + d0 + l;
#pragma unroll
        for (int r = 0; r < 8; ++r) {
            const int i = (r + hi * 8) * 16 + l;
            orow[(size_t)r * DHEAD] = p0[i] + p1[i];
        }
    }
}

extern "C" void kernel_launch(void* const* d_in, const int* in_sizes, int n_in,
                              void* d_out, int out_size, void* d_ws, size_t ws_size,
                              hipStream_t stream) {
    const float* q    = (const float*)d_in[0];
    const float* k    = (const float*)d_in[1];
    const float* v    = (const float*)d_in[2];
    const float* bias = (const float*)d_in[3];

    float* out  = (float*)d_out;                              // B*H*S*D
    float* attn = out + (size_t)2 * 16 * SEQ * DHEAD;         // then B*H*S*S

    dim3 grid(2 * 16 * NTILES);   // 4096 workgroups
    dim3 block(NTHR);
    hipLaunchKernelGGL(attn_fwd_kernel, grid, block, 0, stream,
                       q, k, v, bias, out, attn);
}